// ViTAdapter_89386859364828
// MI455X (gfx1250) — compile-verified
//
#include <hip/hip_runtime.h>
#include <hip/hip_bf16.h>

// ---------------------------------------------------------------------------
// ViT-Adapter deformable-attention injector for MI455X (gfx1250, wave32).
//   D=256, NH=8, DH=32, NL=4, NP=4, B=2, NQ=4096, Nv=87040
// Pipeline:
//   1) value = LN(feat) @ Wv + bv        (WMMA f32 16x16x4, LN fused)
//   2) off   = LN(query) @ Wo + bo       (WMMA)
//   3) aw    = softmax16(LN(query)@Wa+ba)(WMMA + tiny softmax kernel)
//   4) samp  = deformable bilinear gather (wave per (b,q,h), lane=channel)
//   5) out   = query + gamma * (samp @ Wout + bout)   (WMMA, epilogue fused)
// ---------------------------------------------------------------------------

typedef float v2f __attribute__((ext_vector_type(2)));
typedef float v8f __attribute__((ext_vector_type(8)));

namespace cfg {
constexpr int D  = 256;
constexpr int NH = 8;
constexpr int DH = 32;
constexpr int NL = 4;
constexpr int NP = 4;
constexpr int B  = 2;
constexpr int NQ = 4096;              // 64*64
constexpr int NV = 87040;             // 256^2+128^2+64^2+32^2
constexpr int K  = 256;               // GEMM reduction dim (== D)
constexpr int BM = 64;                // rows per block
constexpr int KC = 32;                // K chunk staged in LDS
}  // namespace cfg

// --------------------------- fused LN + WMMA GEMM ---------------------------
// A: [M, 256] row-major (optionally layernormed per row)
// Bw: [256, N] row-major.  D: [M, N].
// MODE 0: D = acc + bias ; MODE 1: D = resid + gamma*(acc + bias)
template <int N, bool LN, int MODE>
__global__ __launch_bounds__(256) void gemm_ln_wmma(
    const float* __restrict__ A, const float* __restrict__ Bw,
    const float* __restrict__ bias,
    const float* __restrict__ lnw, const float* __restrict__ lnb,
    const float* __restrict__ resid, const float* __restrict__ gamma,
    float* __restrict__ Dout) {
  using namespace cfg;
  constexpr int NT = N / 32;        // 16-wide tiles per wave (wave owns N/2 cols)
  constexpr int CG = N / 2;         // column-pair groups per kp row
  constexpr int AIT = (BM * KC / 4) / 256;       // float4 A fetches per thread (2)
  constexpr int BIT = ((KC / 2) * CG) / 256;     // pair fetches per thread (8|4)

  __shared__ __align__(16) float As[BM][KC + 4];        // stride 36: bank-safe
  __shared__ __align__(16) float Bs[KC / 2][N + 16][2]; // k-pair interleaved
  __shared__ float smu[BM], srs[BM];
  __shared__ float slnw[K], slnb[K];

  const int tid  = threadIdx.x;
  const int lane = tid & 31;
  const int wv   = tid >> 5;          // 8 waves
  const int lm   = lane & 15;
  const int lh   = lane >> 4;         // 0/1
  const int m0   = (wv & 3) * 16;     // row tile within block
  const int n0   = (wv >> 2) * (N / 2);
  const int rbase = blockIdx.x * BM;

  if constexpr (LN) {
    slnw[tid] = lnw[tid];
    slnb[tid] = lnb[tid];
    // per-row mean / rstd : wave wv owns rows [wv*8, wv*8+8)
    for (int rr = 0; rr < 8; ++rr) {
      const int row = wv * 8 + rr;
      const float* ap = A + (size_t)(rbase + row) * K;
      float s1 = 0.f, s2 = 0.f;
      for (int c = lane; c < K; c += 32) {
        float x = ap[c];
        s1 += x;
        s2 += x * x;
      }
      for (int sft = 16; sft >= 1; sft >>= 1) {
        s1 += __shfl_xor(s1, sft, 32);
        s2 += __shfl_xor(s2, sft, 32);
      }
      if (lane == 0) {
        const float mu  = s1 * (1.f / 256.f);
        const float var = s2 * (1.f / 256.f) - mu * mu;
        smu[row] = mu;
        srs[row] = rsqrtf(var + 1e-6f);
      }
    }
  }

  // ---- per-thread staging coordinates ----
  // A: idx -> row r = idx/8, col4 c4 = (idx%8)*4    (64 x 32 chunk)
  // B: idx -> kpair kp = idx/CG, col c = 2*(idx%CG) (interleaved pair layout)
  float4 pa[AIT];
  float2 pb0[BIT], pb1[BIT];

  auto fetch = [&](int kc) {
#pragma unroll
    for (int i = 0; i < AIT; ++i) {
      const int idx = i * 256 + tid;
      const int r   = idx >> 3;
      const int c4  = (idx & 7) * 4;
      pa[i] = *reinterpret_cast<const float4*>(
          A + (size_t)(rbase + r) * K + kc * KC + c4);
    }
#pragma unroll
    for (int i = 0; i < BIT; ++i) {
      const int idx = i * 256 + tid;
      const int kp  = idx / CG;
      const int c   = (idx % CG) * 2;
      const float* src = Bw + (size_t)(kc * KC + 2 * kp) * N + c;
      pb0[i] = *reinterpret_cast<const float2*>(src);
      pb1[i] = *reinterpret_cast<const float2*>(src + N);
    }
  };

  fetch(0);

  v8f acc[NT] = {};

#pragma unroll
  for (int kc = 0; kc < K / KC; ++kc) {
    __syncthreads();  // prev compute done reading LDS (kc=0: stats/slnw ready)
    // ---- store staged regs to LDS (apply LN to A here) ----
#pragma unroll
    for (int i = 0; i < AIT; ++i) {
      const int idx = i * 256 + tid;
      const int r   = idx >> 3;
      const int c4  = (idx & 7) * 4;
      float4 o = pa[i];
      if constexpr (LN) {
        const float mu = smu[r], rs = srs[r];
        const int kcol = kc * KC + c4;
        o.x = (o.x - mu) * rs * slnw[kcol + 0] + slnb[kcol + 0];
        o.y = (o.y - mu) * rs * slnw[kcol + 1] + slnb[kcol + 1];
        o.z = (o.z - mu) * rs * slnw[kcol + 2] + slnb[kcol + 2];
        o.w = (o.w - mu) * rs * slnw[kcol + 3] + slnb[kcol + 3];
      }
      *reinterpret_cast<float4*>(&As[r][c4]) = o;
    }
#pragma unroll
    for (int i = 0; i < BIT; ++i) {
      const int idx = i * 256 + tid;
      const int kp  = idx / CG;
      const int c   = (idx % CG) * 2;
      float4 o;  // {B[2kp][c], B[2kp+1][c], B[2kp][c+1], B[2kp+1][c+1]}
      o.x = pb0[i].x;
      o.y = pb1[i].x;
      o.z = pb0[i].y;
      o.w = pb1[i].y;
      *reinterpret_cast<float4*>(&Bs[kp][c][0]) = o;
    }
    // ---- prefetch next chunk while this one computes ----
    if (kc + 1 < K / KC) fetch(kc + 1);
    __syncthreads();

    // ---- WMMA over the chunk : V_WMMA_F32_16X16X4_F32 ----
    // All A fragments of the chunk first (4x ds_load_2addr_b64), then per
    // k-step: all NT B fragments before the 8 back-to-back WMMAs so the
    // XDL pipe runs without per-pair dscnt waits.
    float2 afr[KC / 4];
#pragma unroll
    for (int kk = 0; kk < KC; kk += 4)
      afr[kk >> 2] =
          *reinterpret_cast<const float2*>(&As[m0 + lm][kk + 2 * lh]);

#pragma unroll
    for (int kk = 0; kk < KC; kk += 4) {
      const int kp = (kk >> 1) + lh;
      float2 bfr[NT];
#pragma unroll
      for (int t = 0; t < NT; ++t)
        bfr[t] =
            *reinterpret_cast<const float2*>(&Bs[kp][n0 + t * 16 + lm][0]);
      v2f a;
      a.x = afr[kk >> 2].x;
      a.y = afr[kk >> 2].y;
#pragma unroll
      for (int t = 0; t < NT; ++t) {
        v2f b;
        b.x = bfr[t].x;
        b.y = bfr[t].y;
        acc[t] = __builtin_amdgcn_wmma_f32_16x16x4_f32(
            false, a, false, b, (short)0, acc[t], false, false);
      }
    }
  }

  // ---- epilogue: C/D layout VGPR i -> M = i + 8*lh, N = lm ----
#pragma unroll
  for (int t = 0; t < NT; ++t) {
    const int col = n0 + t * 16 + lm;
    const float bcol = bias[col];
    float gcol = 0.f;
    if constexpr (MODE == 1) gcol = gamma[col];
#pragma unroll
    for (int i = 0; i < 8; ++i) {
      const size_t grow = (size_t)rbase + m0 + i + 8 * lh;
      float v = acc[t][i] + bcol;
      if constexpr (MODE == 1) v = resid[grow * N + col] + gcol * v;
      Dout[grow * N + col] = v;
    }
  }
}

// --------------------------- softmax over 16 --------------------------------
__global__ __launch_bounds__(256) void softmax16(float* __restrict__ aw,
                                                 int total) {
  const int t = blockIdx.x * blockDim.x + threadIdx.x;
  if (t >= total) return;
  float* p = aw + (size_t)t * 16;
  float v[16];
#pragma unroll
  for (int i = 0; i < 16; ++i) v[i] = p[i];
  float m = v[0];
#pragma unroll
  for (int i = 1; i < 16; ++i) m = fmaxf(m, v[i]);
  float s = 0.f;
#pragma unroll
  for (int i = 0; i < 16; ++i) {
    v[i] = __expf(v[i] - m);
    s += v[i];
  }
  const float inv = 1.f / s;
#pragma unroll
  for (int i = 0; i < 16; ++i) p[i] = v[i] * inv;
}

// ----------------------- deformable bilinear sampling -----------------------
// One wave32 per (b, q, h); lane = channel c in [0,32).
// value: [B, Nv, NH, DH]; off: [B*NQ, 256]; aw: [B*NQ*NH, 16] (softmaxed)
// samp: [B*NQ, 256] with col = h*32 + c  (ready as GEMM A matrix)
__global__ __launch_bounds__(256) void msda_sample(
    const float* __restrict__ value, const float* __restrict__ off,
    const float* __restrict__ aw, const float* __restrict__ refp,
    float* __restrict__ samp) {
  using namespace cfg;
  constexpr int LH_[4] = {256, 128, 64, 32};
  constexpr int LW_[4] = {256, 128, 64, 32};
  constexpr int LS_[4] = {0, 65536, 81920, 86016};

  const int w    = (blockIdx.x * blockDim.x + threadIdx.x) >> 5;  // wave id
  const int lane = threadIdx.x & 31;
  const int h    = w & (NH - 1);
  const int bq   = w >> 3;                 // b*NQ + q
  const int q    = bq & (NQ - 1);
  const int b    = bq >> 12;               // NQ = 4096 = 2^12

  const float refx = refp[q * 2 + 0];
  const float refy = refp[q * 2 + 1];

  // lane i < 16 holds the i-th (l,p) offset pair + attention weight
  float oxv = 0.f, oyv = 0.f, awv = 0.f;
  if (lane < 16) {
    const int ob = bq * 256 + h * 32 + lane * 2;
    oxv = off[ob + 0];
    oyv = off[ob + 1];
    awv = aw[(size_t)w * 16 + lane];
  }

  const int c = lane;  // channel
  float acc = 0.f;
#pragma unroll
  for (int j = 0; j < 16; ++j) {
    const int l = j >> 2;
    const float ox = __shfl(oxv, j, 32);
    const float oy = __shfl(oyv, j, 32);
    const float wj = __shfl(awv, j, 32);
    const int Wl = LW_[l], Hl = LH_[l];
    const float px = refx * (float)Wl + ox - 0.5f;
    const float py = refy * (float)Hl + oy - 0.5f;
    const float fx = floorf(px), fy = floorf(py);
    const float wx = px - fx, wy = py - fy;
    const int x0 = (int)fx, y0 = (int)fy;
    const size_t gbase = ((size_t)b * NV + LS_[l]) * 256 + h * 32 + c;
    const float w00 = wj * (1.f - wx) * (1.f - wy);
    const float w01 = wj * wx * (1.f - wy);
    const float w10 = wj * (1.f - wx) * wy;
    const float w11 = wj * wx * wy;
    if (x0 >= 0 && x0 < Wl && y0 >= 0 && y0 < Hl)
      acc += w00 * value[gbase + (size_t)(y0 * Wl + x0) * 256];
    if (x0 + 1 >= 0 && x0 + 1 < Wl && y0 >= 0 && y0 < Hl)
      acc += w01 * value[gbase + (size_t)(y0 * Wl + x0 + 1) * 256];
    if (x0 >= 0 && x0 < Wl && y0 + 1 >= 0 && y0 + 1 < Hl)
      acc += w10 * value[gbase + (size_t)((y0 + 1) * Wl + x0) * 256];
    if (x0 + 1 >= 0 && x0 + 1 < Wl && y0 + 1 >= 0 && y0 + 1 < Hl)
      acc += w11 * value[gbase + (size_t)((y0 + 1) * Wl + x0 + 1) * 256];
  }
  samp[(size_t)bq * 256 + h * 32 + c] = acc;
}

// ------------------------------- launcher -----------------------------------
extern "C" void kernel_launch(void* const* d_in, const int* in_sizes, int n_in,
                              void* d_out, int out_size, void* d_ws,
                              size_t ws_size, hipStream_t stream) {
  using namespace cfg;
  const float* query  = (const float*)d_in[0];   // [B,NQ,256]
  const float* feat   = (const float*)d_in[1];   // [B,NV,256]
  const float* refp   = (const float*)d_in[2];   // [NQ,2]
  // d_in[3]: spatial_shapes (compile-time constants; unused)
  const float* ln_q_w = (const float*)d_in[4];
  const float* ln_q_b = (const float*)d_in[5];
  const float* ln_f_w = (const float*)d_in[6];
  const float* ln_f_b = (const float*)d_in[7];
  const float* Wv     = (const float*)d_in[8];   // [256,256]
  const float* bv     = (const float*)d_in[9];
  const float* Wo     = (const float*)d_in[10];  // [256,256]
  const float* bo     = (const float*)d_in[11];
  const float* Wa     = (const float*)d_in[12];  // [256,128]
  const float* ba     = (const float*)d_in[13];
  const float* Wout   = (const float*)d_in[14];  // [256,256]
  const float* bout   = (const float*)d_in[15];
  const float* gamma  = (const float*)d_in[16];
  float* out = (float*)d_out;

  // workspace layout (fp32): value | off | aw | samp  (~198.5 MB)
  float* ws    = (float*)d_ws;
  float* value = ws;                                    // B*NV*256
  float* offb  = value + (size_t)B * NV * 256;          // B*NQ*256
  float* awb   = offb + (size_t)B * NQ * 256;           // B*NQ*128
  float* samp  = awb + (size_t)B * NQ * 128;            // B*NQ*256

  const int MQ = B * NQ;   // 8192
  const int MV = B * NV;   // 174080

  // 1) value projection, LN(feat) fused
  gemm_ln_wmma<256, true, 0><<<MV / BM, 256, 0, stream>>>(
      feat, Wv, bv, ln_f_w, ln_f_b, nullptr, nullptr, value);
  // 2) sampling offsets, LN(query) fused
  gemm_ln_wmma<256, true, 0><<<MQ / BM, 256, 0, stream>>>(
      query, Wo, bo, ln_q_w, ln_q_b, nullptr, nullptr, offb);
  // 3) attention logits, LN(query) fused
  gemm_ln_wmma<128, true, 0><<<MQ / BM, 256, 0, stream>>>(
      query, Wa, ba, ln_q_w, ln_q_b, nullptr, nullptr, awb);
  // 3b) softmax over NL*NP=16
  softmax16<<<(MQ * NH + 255) / 256, 256, 0, stream>>>(awb, MQ * NH);
  // 4) deformable bilinear sampling: 1 wave per (b,q,h)
  msda_sample<<<(MQ * NH) / 8, 256, 0, stream>>>(value, offb, awb, refp, samp);
  // 5) output projection + residual + gamma
  gemm_ln_wmma<256, false, 1><<<MQ / BM, 256, 0, stream>>>(
      samp, Wout, bout, nullptr, nullptr, query, gamma, out);
}